// SeparableQPProjection_31464930410711
// MI455X (gfx1250) — compile-verified
//
#include <hip/hip_runtime.h>
#include <hip/hip_bf16.h>

#define N_COLS   4096
#define THREADS  256
#define EPT      16      // N_COLS / THREADS
#define NWAVES   (THREADS / 32)
#define ITERS    40

typedef float v2f __attribute__((ext_vector_type(2)));
typedef float v8f __attribute__((ext_vector_type(8)));

// ds_swizzle group-of-32 imm: offset[14:10]=xor_mask, [9:5]=or_mask, [4:0]=and_mask
// SWAPX16: xor=0x10, or=0x00, and=0x1f  -> 0x401F
#define SWIZZLE_SWAPX16 0x401F

// Wave-level sum of 32 lane partials via V_WMMA_F32_16X16X4_F32 with B == ones:
// D[i][j] = sum_k A[i][k]; D's 8 VGPRs cover 8 distinct rows per half-wave, so
// adding the 8 VGPRs then swapping half-waves yields the full 32-lane sum in
// every lane (bit-identical across lanes).
__device__ __forceinline__ float wave_sum_wmma(float partial) {
    v2f a; a.x = partial; a.y = 0.0f;          // one A element per lane, rest 0
    v2f b; b.x = 1.0f;    b.y = 1.0f;          // B = all ones (layout-agnostic)
    v8f c = {0.f, 0.f, 0.f, 0.f, 0.f, 0.f, 0.f, 0.f};
    v8f d = __builtin_amdgcn_wmma_f32_16x16x4_f32(
        /*neg_a=*/false, a, /*neg_b=*/false, b,
        /*c_mod=*/(short)0, c, /*reuse_a=*/false, /*reuse_b=*/false);
    float t = ((d[0] + d[1]) + (d[2] + d[3])) + ((d[4] + d[5]) + (d[6] + d[7]));
    t += __int_as_float(__builtin_amdgcn_ds_swizzle(__float_as_int(t),
                                                    SWIZZLE_SWAPX16));
    return t;
}

__global__ __launch_bounds__(THREADS) void qp_bisect_kernel(
    const float* __restrict__ z, const float* __restrict__ gamma,
    const float* __restrict__ m, const float* __restrict__ M,
    const float* __restrict__ xi, float* __restrict__ out) {
    // Double-buffered wave-sum slots: one barrier per bisection iteration.
    __shared__ __align__(16) float s_red[2][NWAVES];
    __shared__ float s_min[NWAVES];
    __shared__ float s_max[NWAVES];

    const int t    = threadIdx.x;
    const int lane = t & 31;
    const int wid  = t >> 5;
    const int row  = blockIdx.x;
    const long base  = (long)row * N_COLS + t * EPT;
    const int  cbase = t * EPT;

    // Row tile resident in registers across all 40 iterations.
    float zv[EPT], igv[EPT], mv[EPT], Mv[EPT];

    float lo =  3.402823466e38f;
    float hi = -3.402823466e38f;

    #pragma unroll
    for (int q = 0; q < EPT / 4; ++q) {
        const int k = q * 4;
        const float4 z4 = ((const float4*)(z + base))[q];
        const float4 g4 = ((const float4*)(gamma + cbase))[q];
        const float4 m4 = ((const float4*)(m + cbase))[q];
        const float4 M4 = ((const float4*)(M + cbase))[q];
        const float zz[4] = {z4.x, z4.y, z4.z, z4.w};
        const float gg[4] = {g4.x, g4.y, g4.z, g4.w};
        const float mm[4] = {m4.x, m4.y, m4.z, m4.w};
        const float MM[4] = {M4.x, M4.y, M4.z, M4.w};
        #pragma unroll
        for (int j = 0; j < 4; ++j) {
            const float tg = 2.0f * gg[j];
            zv[k + j]  = zz[j];
            mv[k + j]  = mm[j];
            Mv[k + j]  = MM[j];
            igv[k + j] = 1.0f / tg;                 // hoisted out of bisection
            lo = fminf(lo, tg * (mm[j] - zz[j]));   // lam_low candidates
            hi = fmaxf(hi, tg * (MM[j] - zz[j]));   // lam_high candidates
        }
    }

    // One-time block min/max reduction for the bisection bracket.
    #pragma unroll
    for (int o = 16; o > 0; o >>= 1) {
        lo = fminf(lo, __shfl_xor(lo, o, 32));
        hi = fmaxf(hi, __shfl_xor(hi, o, 32));
    }
    if (lane == 0) { s_min[wid] = lo; s_max[wid] = hi; }
    __syncthreads();
    lo = s_min[0]; hi = s_max[0];
    #pragma unroll
    for (int i = 1; i < NWAVES; ++i) {
        lo = fminf(lo, s_min[i]);
        hi = fmaxf(hi, s_max[i]);
    }
    __syncthreads();

    const float xi_row = xi[row];

    // 40 bisection iterations, entirely register/LDS resident.
    for (int it = 0; it < ITERS; ++it) {
        const float mid = 0.5f * (lo + hi);
        float p0 = 0.f, p1 = 0.f;                   // two chains for ILP
        #pragma unroll
        for (int k = 0; k < EPT; k += 2) {
            p0 += fminf(Mv[k + 0], fmaxf(mv[k + 0], fmaf(mid, igv[k + 0], zv[k + 0])));
            p1 += fminf(Mv[k + 1], fmaxf(mv[k + 1], fmaf(mid, igv[k + 1], zv[k + 1])));
        }
        const float wsum = wave_sum_wmma(p0 + p1);
        const int buf = it & 1;
        if (lane == 0) s_red[buf][wid] = wsum;
        __syncthreads();
        const float4 r0 = ((const float4*)&s_red[buf][0])[0];
        const float4 r1 = ((const float4*)&s_red[buf][0])[1];
        const float g = (((r0.x + r0.y) + (r0.z + r0.w)) +
                         ((r1.x + r1.y) + (r1.z + r1.w))) - xi_row;
        const bool pos = g > 0.f;
        lo = pos ? lo : mid;    // g <= 0 -> raise lower bound
        hi = pos ? mid : hi;    // g  > 0 -> lower upper bound
        // next iteration writes the other buffer; its barrier orders reuse
    }

    const float lam = 0.5f * (lo + hi);
    #pragma unroll
    for (int q = 0; q < EPT / 4; ++q) {
        const int k = q * 4;
        float4 o4;
        o4.x = fminf(Mv[k + 0], fmaxf(mv[k + 0], fmaf(lam, igv[k + 0], zv[k + 0])));
        o4.y = fminf(Mv[k + 1], fmaxf(mv[k + 1], fmaf(lam, igv[k + 1], zv[k + 1])));
        o4.z = fminf(Mv[k + 2], fmaxf(mv[k + 2], fmaf(lam, igv[k + 2], zv[k + 2])));
        o4.w = fminf(Mv[k + 3], fmaxf(mv[k + 3], fmaf(lam, igv[k + 3], zv[k + 3])));
        ((float4*)(out + base))[q] = o4;
    }
}

extern "C" void kernel_launch(void* const* d_in, const int* in_sizes, int n_in,
                              void* d_out, int out_size, void* d_ws, size_t ws_size,
                              hipStream_t stream) {
    const float* z     = (const float*)d_in[0];
    const float* gamma = (const float*)d_in[1];
    const float* m     = (const float*)d_in[2];
    const float* M     = (const float*)d_in[3];
    const float* xi    = (const float*)d_in[4];
    float* out = (float*)d_out;
    const int rows = in_sizes[4];   // B
    qp_bisect_kernel<<<dim3(rows), dim3(THREADS), 0, stream>>>(z, gamma, m, M, xi, out);
}